// MOELayer_3521873183218
// MI455X (gfx1250) — compile-verified
//
#include <hip/hip_runtime.h>
#include <hip/hip_bf16.h>

// ---- problem constants (match reference) ----
#define NUM_EXPERT 64
#define D_IN  512
#define D_OUT 512
#define N_TOK 65536

// ---- tiling ----
#define TILE_M 32          // tokens per block
#define KC     16          // K chunk staged in LDS
#define APITCH 20          // LDS row pitch (floats): 16B-aligned rows, conflict-free b64 reads
#define BPITCH 20          // matches TDM pad: 16 dwords data + 4 dwords pad
#define NCHUNK (D_IN / KC) // 32

typedef __attribute__((ext_vector_type(2))) float v2f;
typedef __attribute__((ext_vector_type(4))) float v4f;
typedef __attribute__((ext_vector_type(8))) float v8f;
typedef __attribute__((ext_vector_type(4))) unsigned int u32x4;
typedef __attribute__((ext_vector_type(8))) int i32x8;
typedef __attribute__((ext_vector_type(4))) int i32x4;

// workspace layout (ints)
#define WS_COUNTS  0
#define WS_CURSORS 64
#define WS_TOKOFF  128
#define WS_TILEOFF 256
#define WS_SORTED  384

__global__ void moe_init(int* ws) {
    if (threadIdx.x < 128) ws[threadIdx.x] = 0;
}

__global__ void moe_hist(const int* __restrict__ gate, int* __restrict__ ws, int n) {
    int i = blockIdx.x * blockDim.x + threadIdx.x;
    if (i < n) atomicAdd(&ws[WS_COUNTS + gate[i]], 1);
}

__global__ void moe_scan(int* __restrict__ ws) {
    if (threadIdx.x == 0 && blockIdx.x == 0) {
        int a = 0, b = 0;
        for (int e = 0; e < NUM_EXPERT; ++e) {
            ws[WS_TOKOFF + e]  = a;
            ws[WS_TILEOFF + e] = b;
            int c = ws[WS_COUNTS + e];
            a += c;
            b += (c + TILE_M - 1) / TILE_M;
        }
        ws[WS_TOKOFF + NUM_EXPERT]  = a;
        ws[WS_TILEOFF + NUM_EXPERT] = b;
    }
}

__global__ void moe_scatter(const int* __restrict__ gate, int* __restrict__ ws, int n) {
    int i = blockIdx.x * blockDim.x + threadIdx.x;
    if (i < n) {
        int g = gate[i];
        int pos = ws[WS_TOKOFF + g] + atomicAdd(&ws[WS_CURSORS + g], 1);
        ws[WS_SORTED + pos] = i;
    }
}

// ---- TDM: DMA one B chunk W[e][0..511][k0..k0+15] -> LDS with 20-float padded pitch ----
// D# per CDNA5 ISA ch.8: group0 = {flags, lds_addr, global_addr, type}, group1 = dims/strides.
__device__ __forceinline__ void tdm_load_b_chunk(const float* gsrc, unsigned lds_off) {
    unsigned long long ga = (unsigned long long)(uintptr_t)gsrc;
    u32x4 g0;
    g0[0] = 1u;                                   // count=1 (valid), no gather, no restore
    g0[1] = lds_off;                              // lds_addr (bytes, wave-relative)
    g0[2] = (unsigned)(ga & 0xffffffffu);         // global_addr[31:0]
    g0[3] = (unsigned)((ga >> 32) & 0x01ffffffu)  // global_addr[56:32]
          | (2u << 30);                           // type=2 ("image")
    i32x8 g1;
    g1[0] = (2 << 16)      // data_size = 4 bytes
          | (1 << 20)      // pad_enable
          | (3 << 22)      // pad_interval: 16 dwords between pads
          | (3 << 25);     // pad_amount: 4 dwords  -> 20-float LDS pitch
    g1[1] = (int)(512u << 16);  // tensor_dim0[15:0]=512 in [31:16]; atomic_barrier_addr=0
    g1[2] = (int)(512u << 16);  // tensor_dim0[31:16]=0; tensor_dim1[15:0]=512
    g1[3] = (int)(16u  << 16);  // tensor_dim1[31:16]=0; tile_dim0=16
    g1[4] = 512;                // tile_dim1=512 rows; tile_dim2=0
    g1[5] = 512;                // tensor_dim0_stride[31:0]=512 elements
    g1[6] = 0;                  // stride0 hi / stride1 lo = 0
    g1[7] = 0;
    i32x4 z4 = {0, 0, 0, 0};    // groups 2/3 unused (2D tensor)
    i32x8 z8 = {0, 0, 0, 0, 0, 0, 0, 0};
    __builtin_amdgcn_tensor_load_to_lds(g0, g1, z4, z4, z8, 0);
}

// ---- stage gathered A chunk: 32 token rows x 16 floats ----
__device__ __forceinline__ void stage_a(const float* __restrict__ inp, const int* rowTok,
                                        float* As, int k0, int tid) {
    if (tid < 128) {
        int r  = tid >> 2;
        int kq = (tid & 3) << 2;
        int tok = rowTok[r];
        if (tok < 0) tok = rowTok[0];   // padding rows: valid address, values never stored
        v4f a = __builtin_nontemporal_load(
            (const v4f*)(inp + (size_t)tok * D_IN + k0 + kq));
        *(v4f*)(&As[r * APITCH + kq]) = a;
    }
}

// Grouped GEMM: block = one 32-token tile of one expert, all 512 outputs.
// 8 wave32; wave w owns output columns [w*64, w*64+64). TDM double-buffers B.
__global__ __launch_bounds__(256) void moe_gemm(
    const float* __restrict__ inp,      // [N_TOK, D_IN]
    const float* __restrict__ weight,   // [NUM_EXPERT, D_OUT, D_IN]
    const int*   __restrict__ ws,
    float*       __restrict__ out)      // [N_TOK, D_OUT]
{
    __shared__ float As[2][TILE_M * APITCH];   //  5 KB
    __shared__ float Bs[2][D_OUT * BPITCH];    // 80 KB (TDM destination)
    __shared__ int   rowTok[TILE_M];
    __shared__ int   sTileOff[NUM_EXPERT + 1];

    const int tid = threadIdx.x;
    const int t   = blockIdx.x;

    const int* tokOff  = ws + WS_TOKOFF;
    const int* tileOff = ws + WS_TILEOFF;
    const int* sorted  = ws + WS_SORTED;

    if (tid < NUM_EXPERT + 1) sTileOff[tid] = tileOff[tid];
    __syncthreads();

    if (t >= sTileOff[NUM_EXPERT]) return;   // block-uniform exit

    int e = 0;
    while (sTileOff[e + 1] <= t) ++e;

    const int row0   = tokOff[e] + (t - sTileOff[e]) * TILE_M;
    const int tokEnd = tokOff[e + 1];

    if (tid < TILE_M) {
        int gr = row0 + tid;
        rowTok[tid] = (gr < tokEnd) ? sorted[gr] : -1;
    }
    __syncthreads();                          // rowTok ready for A staging

    const int lane     = tid & 31;
    const int wave     = tid >> 5;
    const int laneHalf = lane >> 4;           // 0: lanes 0-15, 1: lanes 16-31
    const int ln       = lane & 15;

    const float* Wb = weight + (size_t)e * (D_OUT * D_IN);

    v8f acc[2][4] = {};                       // 2 M-tiles x 4 N-tiles, 16x16 each

    // ---- prologue: stage chunk 0 into buffer 0 ----
    if (tid < 32) tdm_load_b_chunk(Wb, (unsigned)(uintptr_t)&Bs[0][0]);
    stage_a(inp, rowTok, &As[0][0], 0, tid);

    int buf = 0;
    for (int i = 0; i < NCHUNK; ++i) {
        if (tid < 32) __builtin_amdgcn_s_wait_tensorcnt(0);  // chunk i DMA complete
        __syncthreads();   // chunk i visible to all waves; buffer buf^1 free again

        const int nbuf = buf ^ 1;
        const int k1   = (i + 1) * KC;
        if (k1 < D_IN) {   // prefetch chunk i+1, overlapped with WMMA below
            if (tid < 32) tdm_load_b_chunk(Wb + k1, (unsigned)(uintptr_t)&Bs[nbuf][0]);
            stage_a(inp, rowTok, &As[nbuf][0], k1, tid);
        }

        const float* Ab = &As[buf][0];
        const float* Bb = &Bs[buf][0];
#pragma unroll
        for (int kk = 0; kk < KC; kk += 4) {
            const int kb = kk + (laneHalf << 1);  // lanes 0-15: K=kk,kk+1; 16-31: kk+2,kk+3
            v2f a0 = *(const v2f*)(&Ab[(ln)      * APITCH + kb]);
            v2f a1 = *(const v2f*)(&Ab[(16 + ln) * APITCH + kb]);
#pragma unroll
            for (int nt = 0; nt < 4; ++nt) {
                int o = wave * 64 + nt * 16 + ln;
                v2f b = *(const v2f*)(&Bb[o * BPITCH + kb]);
                acc[0][nt] = __builtin_amdgcn_wmma_f32_16x16x4_f32(
                    false, a0, false, b, (short)0, acc[0][nt], false, false);
                acc[1][nt] = __builtin_amdgcn_wmma_f32_16x16x4_f32(
                    false, a1, false, b, (short)0, acc[1][nt], false, false);
            }
        }
        buf = nbuf;
    }

    // ---- scatter-store C: VGPR r holds M=r (lanes 0-15) / M=r+8 (lanes 16-31), N=ln ----
#pragma unroll
    for (int mt = 0; mt < 2; ++mt) {
#pragma unroll
        for (int nt = 0; nt < 4; ++nt) {
#pragma unroll
            for (int r = 0; r < 8; ++r) {
                int m = mt * 16 + laneHalf * 8 + r;
                int tok = rowTok[m];
                if (tok >= 0) {
                    __builtin_nontemporal_store(
                        acc[mt][nt][r],
                        out + (size_t)tok * D_OUT + wave * 64 + nt * 16 + ln);
                }
            }
        }
    }
}

extern "C" void kernel_launch(void* const* d_in, const int* in_sizes, int n_in,
                              void* d_out, int out_size, void* d_ws, size_t ws_size,
                              hipStream_t stream) {
    const float* inp    = (const float*)d_in[0];
    const int*   gate   = (const int*)d_in[1];
    const float* weight = (const float*)d_in[2];
    float*       out    = (float*)d_out;
    int*         ws     = (int*)d_ws;

    (void)in_sizes; (void)n_in; (void)out_size; (void)ws_size;

    moe_init<<<1, 128, 0, stream>>>(ws);
    moe_hist<<<N_TOK / 256, 256, 0, stream>>>(gate, ws, N_TOK);
    moe_scan<<<1, 32, 0, stream>>>(ws);
    moe_scatter<<<N_TOK / 256, 256, 0, stream>>>(gate, ws, N_TOK);

    const int maxTiles = N_TOK / TILE_M + NUM_EXPERT;   // 2112
    moe_gemm<<<maxTiles, 256, 0, stream>>>(inp, weight, ws, out);
}